// GradTree_54322746360310
// MI455X (gfx1250) — compile-verified
//
#include <hip/hip_runtime.h>
#include <stdint.h>

#define N_ROWS   (1024 * 255)   // trees * nodes
#define N_FEAT   512
#define NTHREADS 256

typedef __attribute__((ext_vector_type(2)))  float    v2f;
typedef __attribute__((ext_vector_type(8)))  float    v8f;
typedef __attribute__((ext_vector_type(4)))  uint32_t v4u;
typedef __attribute__((ext_vector_type(8)))  int      v8i;
typedef __attribute__((ext_vector_type(4)))  int      v4i;

#if defined(__gfx1250__) && __has_builtin(__builtin_amdgcn_tensor_load_to_lds) && \
    __has_builtin(__builtin_amdgcn_s_wait_tensorcnt)
#define USE_TDM 1
#else
#define USE_TDM 0
#endif

#if defined(__gfx1250__) && __has_builtin(__builtin_amdgcn_wmma_f32_16x16x4_f32)
#define USE_WMMA 1
#else
#define USE_WMMA 0
#endif

__global__ void __launch_bounds__(NTHREADS)
entmax15_kernel(const float* __restrict__ logits, float* __restrict__ out)
{
    __shared__ float s_x[N_FEAT];     // raw row -> x = v - max -> sorted x (descending)
    __shared__ float s_cs[N_FEAT];    // cumsum of sorted x
    __shared__ float s_cq[N_FEAT];    // cumsum of sorted x^2
    __shared__ float s_tau[N_FEAT];   // per-prefix tau
    __shared__ float s_red[NTHREADS]; // max-reduction scratch
    __shared__ int   s_support;

    const int tid = threadIdx.x;
    const int row = blockIdx.x;
    const float* __restrict__ src = logits + (size_t)row * N_FEAT;

    if (tid == 0) s_support = 0;

    // ---- Stage 1: row -> LDS via Tensor Data Mover (wave 0 issues the DMA) ----
#if USE_TDM
    if (tid < 32) {  // wave-uniform branch: only wave 0
        const uint64_t ga  = (uint64_t)(uintptr_t)src;
        const uint32_t lds = (uint32_t)(uintptr_t)(void*)&s_x[0]; // flat low 32b == LDS byte offset

        // D# group 0: count=1 (valid), lds_addr, global_addr[56:0], type=2 ("image")
        v4u g0;
        g0.x = 1u;
        g0.y = lds;
        g0.z = (uint32_t)ga;
        g0.w = ((uint32_t)((ga >> 32) & 0x01FFFFFFull)) | (2u << 30);

        // D# group 1: mask=0, data_size=2 (4B), dim0=512, dim1=1, tile_dim0=512,
        //             tensor_dim0_stride=512, everything else 0
        v8i g1;
        g1[0] = (int)(2u << 16);     // data_size=4B at [17:16]
        g1[1] = (int)(512u << 16);   // tensor_dim0[15:0] at bits [63:48]
        g1[2] = (int)(1u << 16);     // tensor_dim1[15:0]=1 at bits [95:80]
        g1[3] = (int)(512u << 16);   // tile_dim0=512 at bits [127:112]
        g1[4] = 0;                   // tile_dim1=0 (unused), tile_dim2=0
        g1[5] = 512;                 // tensor_dim0_stride[31:0]
        g1[6] = 0;
        g1[7] = 0;

        v4i z4 = {0, 0, 0, 0};                   // groups 2/3 unused (<=2D tensor)
        v8i z8 = {0, 0, 0, 0, 0, 0, 0, 0};
        __builtin_amdgcn_tensor_load_to_lds(g0, g1, z4, z4, z8, 0);
        __builtin_amdgcn_s_wait_tensorcnt(0);
    }
#else
    s_x[tid]       = src[tid];
    s_x[tid + 256] = src[tid + 256];
#endif
    __syncthreads();

    // ---- Stage 2: x = logits/2 - max ----
    float v0 = s_x[tid]       * 0.5f;
    float v1 = s_x[tid + 256] * 0.5f;
    s_red[tid] = fmaxf(v0, v1);
    __syncthreads();
    for (int s = NTHREADS / 2; s > 0; s >>= 1) {
        if (tid < s) s_red[tid] = fmaxf(s_red[tid], s_red[tid + s]);
        __syncthreads();
    }
    const float vmax = s_red[0];
    const float x0 = v0 - vmax;
    const float x1 = v1 - vmax;
    __syncthreads();                 // everyone done reading s_x / s_red
    s_x[tid]       = x0;
    s_x[tid + 256] = x1;
    __syncthreads();

    // ---- Stage 3: bitonic sort, descending (45 stages) ----
    for (int k = 2; k <= N_FEAT; k <<= 1) {
        for (int j = k >> 1; j > 0; j >>= 1) {
            const int i = ((tid & ~(j - 1)) << 1) | (tid & (j - 1));
            const int p = i | j;
            const float a = s_x[i];
            const float b = s_x[p];
            const bool desc = ((i & k) == 0);     // final merge (k=512) -> all descending
            const bool sw = desc ? (a < b) : (a > b);
            if (sw) { s_x[i] = b; s_x[p] = a; }
            __syncthreads();
        }
    }

    // ---- Stage 4: inclusive scans of xs and xs^2 on the matrix pipe ----
    // W = R(16x16 segs x elems) x U(upper-tri ones) via 4 chained K=4 fp32 WMMAs.
    // wave 0 scans x, wave 1 scans x^2; cross-segment offsets via wave32 shfl scan.
    const int wave = tid >> 5;
    const int lane = tid & 31;
    if (wave < 2) {
        const bool sq = (wave == 1);
        float* dst = sq ? s_cq : s_cs;
#if USE_WMMA
        const int m  = lane & 15;   // A-matrix row (segment) and D column (element)
        const int kh = lane >> 4;   // K half: VGPR0 holds K0/K2, VGPR1 holds K1/K3
        for (int t = 0; t < 2; ++t) {
            const int segBase = t << 4;
            v8f acc = {};
#pragma unroll
            for (int c = 0; c < 4; ++c) {
                const int kb = 4 * c + (kh ? 2 : 0);
                float a0 = s_x[(segBase + m) * 16 + kb];
                float a1 = s_x[(segBase + m) * 16 + kb + 1];
                if (sq) { a0 *= a0; a1 *= a1; }
                v2f A = { a0, a1 };
                // B = U[4c+k][n], n = lane&15 : ones where K <= N (inclusive scan)
                v2f B = { (kb     <= m) ? 1.0f : 0.0f,
                          (kb + 1 <= m) ? 1.0f : 0.0f };
                acc = __builtin_amdgcn_wmma_f32_16x16x4_f32(
                          false, A, false, B, (short)0, acc, false, false);
            }
#pragma unroll
            for (int v = 0; v < 8; ++v) {        // D: VGPR v -> M=v (lanes 0-15) / v+8
                const int seg = segBase + v + (kh ? 8 : 0);
                dst[seg * 16 + m] = acc[v];
            }
        }
#else
        float run = 0.0f;
        for (int e = 0; e < 16; ++e) {
            float vv = s_x[lane * 16 + e];
            if (sq) vv *= vv;
            run += vv;
            dst[lane * 16 + e] = run;
        }
#endif
        __builtin_amdgcn_wave_barrier();          // order LDS writes vs. reads (same wave)
        // exclusive scan of the 32 segment totals across lanes
        const float tot = dst[lane * 16 + 15];
        float inc = tot;
        for (int d = 1; d < 32; d <<= 1) {
            float y = __shfl_up(inc, d, 32);
            if (lane >= d) inc += y;
        }
        const float off = inc - tot;
        for (int e = 0; e < 16; ++e) dst[lane * 16 + e] += off;
    }
    __syncthreads();

    // ---- Stage 5: tau per prefix, support count, tau* ----
    int cnt = 0;
#pragma unroll
    for (int q = 0; q < 2; ++q) {
        const int p = tid + q * 256;
        const float rho  = (float)(p + 1);
        const float mean = s_cs[p] / rho;
        const float msq  = s_cq[p] / rho;
        const float delta = (1.0f - rho * (msq - mean * mean)) / rho;
        const float tau = mean - sqrtf(fmaxf(delta, 0.0f));
        s_tau[p] = tau;
        cnt += (tau <= s_x[p]) ? 1 : 0;
    }
    atomicAdd(&s_support, cnt);
    __syncthreads();
    const float tstar = s_tau[s_support - 1];

    // ---- Stage 6: outputs at original positions ----
    float* __restrict__ drow = out + (size_t)row * N_FEAT;
    const float r0 = fmaxf(x0 - tstar, 0.0f);
    const float r1 = fmaxf(x1 - tstar, 0.0f);
    drow[tid]       = r0 * r0;
    drow[tid + 256] = r1 * r1;
}

extern "C" void kernel_launch(void* const* d_in, const int* in_sizes, int n_in,
                              void* d_out, int out_size, void* d_ws, size_t ws_size,
                              hipStream_t stream) {
    (void)in_sizes; (void)n_in; (void)out_size; (void)d_ws; (void)ws_size;
    const float* logits = (const float*)d_in[0];
    float* out = (float*)d_out;
    hipLaunchKernelGGL(entmax15_kernel, dim3(N_ROWS), dim3(NTHREADS), 0, stream,
                       logits, out);
}